// GraphVAE_43782896616017
// MI455X (gfx1250) — compile-verified
//
#include <hip/hip_runtime.h>
#include <hip/hip_bf16.h>

// ---------------- problem constants (from reference) ----------------
#define N_GRAPHS 256
#define N_PER    256
#define NN       (N_GRAPHS * N_PER)   // 65536 nodes
#define DEG      16
#define EE       (NN * DEG)           // 1048576 edges
#define IN_FEATS 16
#define HIDDEN   64
#define Z_DIM    128
#define DEC_IN   (3 * HIDDEN + IN_FEATS)  // 208

typedef float v2f __attribute__((ext_vector_type(2)));
typedef float v8f __attribute__((ext_vector_type(8)));
typedef unsigned int u32x4 __attribute__((ext_vector_type(4)));
typedef int i32x8 __attribute__((ext_vector_type(8)));
typedef int i32x4 __attribute__((ext_vector_type(4)));

#if defined(__has_builtin)
#if __has_builtin(__builtin_amdgcn_tensor_load_to_lds) && __has_builtin(__builtin_amdgcn_s_wait_tensorcnt)
#define HAVE_TDM 1
#endif
#endif

// ---------------- CSR construction + norms ----------------

__global__ void k_zero_i(int* __restrict__ p, long long n) {
  long long i = (long long)blockIdx.x * blockDim.x + threadIdx.x;
  long long stride = (long long)gridDim.x * blockDim.x;
  for (; i < n; i += stride) p[i] = 0;
}

__global__ void k_degree(const int* __restrict__ src, const int* __restrict__ dst,
                         int* __restrict__ deg_src, int* __restrict__ deg_dst) {
  int e = blockIdx.x * blockDim.x + threadIdx.x;
  if (e < EE) {
    atomicAdd(&deg_src[src[e]], 1);
    atomicAdd(&deg_dst[dst[e]], 1);
  }
}

__global__ void k_finalize_norm(const int* __restrict__ deg_src, const int* __restrict__ deg_dst,
                                float* __restrict__ ns, float* __restrict__ nd) {
  int i = blockIdx.x * blockDim.x + threadIdx.x;
  if (i < NN) {
    ns[i] = rsqrtf(fmaxf((float)deg_src[i], 1.0f));   // D_out^{-1/2}
    nd[i] = rsqrtf(fmaxf((float)deg_dst[i], 1.0f));   // D_in^{-1/2}
  }
}

// Exclusive prefix-sum of deg_dst[NN] -> row_start[NN+1]. One block, 1024 thr x 64 elems.
__global__ void k_scan(const int* __restrict__ deg, int* __restrict__ row_start) {
  __shared__ int partial[1024];
  int t = threadIdx.x;
  int base = t * 64;
  int s = 0;
#pragma unroll 4
  for (int j = 0; j < 64; ++j) s += deg[base + j];
  partial[t] = s;
  __syncthreads();
  for (int off = 1; off < 1024; off <<= 1) {   // Hillis-Steele inclusive scan
    int v = (t >= off) ? partial[t - off] : 0;
    __syncthreads();
    partial[t] += v;
    __syncthreads();
  }
  int acc = (t == 0) ? 0 : partial[t - 1];     // exclusive run offset
  for (int j = 0; j < 64; ++j) { row_start[base + j] = acc; acc += deg[base + j]; }
  if (t == 1023) row_start[NN] = acc;          // == EE
}

__global__ void k_fill(const int* __restrict__ src, const int* __restrict__ dst,
                       const int* __restrict__ row_start, int* __restrict__ cursor,
                       int* __restrict__ nbr) {
  int e = blockIdx.x * blockDim.x + threadIdx.x;
  if (e < EE) {
    int d = dst[e];
    int pos = atomicAdd(&cursor[d], 1);
    nbr[row_start[d] + pos] = src[e];
  }
}

// ---------------- gather-based aggregation (no atomics, L2-resident) ----------------
__global__ void k_gather(const float* __restrict__ x, const int* __restrict__ row_start,
                         const int* __restrict__ nbr, const float* __restrict__ ns,
                         const float* __restrict__ nd, float* __restrict__ agg,
                         int din, long long total) {
  long long t = (long long)blockIdx.x * blockDim.x + threadIdx.x;
  if (t >= total) return;
  int i = (int)(t / din);
  int f = (int)(t - (long long)i * din);
  int b0 = row_start[i], b1 = row_start[i + 1];
  float acc = 0.0f;
  for (int j = b0; j < b1; ++j) {
    int s = nbr[j];
    acc += x[(long long)s * din + f] * ns[s];
  }
  agg[t] = acc * nd[i];
}

// Gathered features are the virtual concat [latent(64)|raw(16)|h1(64)|h3(64)] of the src node.
__global__ void k_gather_concat(const float* __restrict__ latent, const float* __restrict__ raw,
                                const float* __restrict__ h1, const float* __restrict__ h3,
                                const int* __restrict__ row_start, const int* __restrict__ nbr,
                                const float* __restrict__ ns, const float* __restrict__ nd,
                                float* __restrict__ agg, long long total) {
  long long t = (long long)blockIdx.x * blockDim.x + threadIdx.x;
  if (t >= total) return;
  int i = (int)(t / DEC_IN);
  int f = (int)(t - (long long)i * DEC_IN);
  const float* srcbuf;
  int srcdim, fo;
  if (f < HIDDEN)                     { srcbuf = latent; srcdim = HIDDEN;   fo = f; }
  else if (f < HIDDEN + IN_FEATS)     { srcbuf = raw;    srcdim = IN_FEATS; fo = f - HIDDEN; }
  else if (f < 2 * HIDDEN + IN_FEATS) { srcbuf = h1;     srcdim = HIDDEN;   fo = f - HIDDEN - IN_FEATS; }
  else                                { srcbuf = h3;     srcdim = HIDDEN;   fo = f - 2 * HIDDEN - IN_FEATS; }
  int b0 = row_start[i], b1 = row_start[i + 1];
  float acc = 0.0f;
  for (int j = b0; j < b1; ++j) {
    int s = nbr[j];
    acc += srcbuf[(long long)s * srcdim + fo] * ns[s];
  }
  agg[t] = acc * nd[i];
}

// ---------------- small elementwise kernels ----------------

__global__ void k_reparam(const float* __restrict__ mu, const float* __restrict__ lv,
                          const float* __restrict__ eps, float* __restrict__ z) {
  int i = blockIdx.x * blockDim.x + threadIdx.x;
  if (i < N_GRAPHS * Z_DIM) z[i] = mu[i] + eps[i] * __expf(0.5f * lv[i]);
}

__global__ void k_fixcol0(const float* __restrict__ raw, float* __restrict__ recon) {
  int i = blockIdx.x * blockDim.x + threadIdx.x;
  if (i < NN) recon[(long long)i * IN_FEATS] = raw[(long long)i * IN_FEATS];
}

// ---------------- TDM helper: async 2D tile load (rows x 64 f32) to LDS ----------------
#ifdef HAVE_TDM
// D# packing per CDNA5 ISA (group0: count/lds/global/type,
// group1: data_size, tensor dims/strides, tile dims). 2D tensor -> groups 2/3 zero.
// This toolchain's builtin is the 6-arg form:
//   (uint32x4 g0, int32x8 g1, int32x4 g2, int32x4 g3, int32x8 g4, i32 cpol)
__device__ __forceinline__ void tdm_load_panel(const float* gptr, unsigned lds_off,
                                               int rows, int Nc) {
  unsigned long long ga = (unsigned long long)(uintptr_t)gptr;
  u32x4 g0;
  g0.x = 1u;                                                 // count=1 (valid user D#)
  g0.y = lds_off;                                            // LDS byte address
  g0.z = (unsigned)(ga & 0xFFFFFFFFu);                       // global_addr[31:0]
  g0.w = (unsigned)((ga >> 32) & 0x01FFFFFFu) | (2u << 30);  // global_addr[56:32] | type=2
  unsigned td0 = 0x40000000u, td1 = 0x40000000u;             // huge tensor dims: no OOB clip
  unsigned long long st0 = (unsigned long long)(unsigned)Nc; // dim0 stride (elements)
  i32x8 g1;
  g1[0] = (int)(2u << 16);                                   // wg_mask=0, data_size=2 (4B)
  g1[1] = (int)((td0 & 0xFFFFu) << 16);                      // tensor_dim0[15:0] @ bit48
  g1[2] = (int)(((td0 >> 16) & 0xFFFFu) | ((td1 & 0xFFFFu) << 16));
  g1[3] = (int)(((td1 >> 16) & 0xFFFFu) | (64u << 16));      // tile_dim0=64 @ bits127:112
  g1[4] = (int)((unsigned)rows & 0xFFFFu);                   // tile_dim1=rows, tile_dim2=0
  g1[5] = (int)(st0 & 0xFFFFFFFFu);                          // tensor_dim0_stride[31:0]
  g1[6] = (int)((st0 >> 32) & 0xFFFFu);                      // stride[47:32] | dim1_stride lo=0
  g1[7] = 0;
  i32x4 g2 = {0, 0, 0, 0};
  i32x4 g3 = {0, 0, 0, 0};
  i32x8 g4 = {0, 0, 0, 0, 0, 0, 0, 0};
  __builtin_amdgcn_tensor_load_to_lds(g0, g1, g2, g3, g4, 0);
}
#endif

// ---------------- WMMA f32 GEMM, TDM-staged B panels, 128x64 macro-tile per block ------
// out = act(A @ B + bias); A:[M,K] rm, B:[K,Nc] rm, bias:[Nc].
// Requires M%128==0, Nc%64==0, K%4==0. Block = 256 thr = 8 waves.
// B k-panels (KBLK x 64, row-major [k][64]) double-buffered in LDS via TENSOR_LOAD_TO_LDS
// issued by wave 0 for panel p+1 while all waves run V_WMMA_F32_16X16X4_F32 on panel p.
#define KBLK 32

__global__ void k_wmma_gemm_lds(const float* __restrict__ A, const float* __restrict__ B,
                                const float* __restrict__ bias, float* __restrict__ C,
                                int M, int K, int Nc, int relu) {
  __shared__ float Bs[2][KBLK * 64];         // 2 x 8 KB panels

  int w    = threadIdx.x >> 5;               // wave in block: 0..7
  int lane = threadIdx.x & 31;
  int half = lane >> 4;                      // K-slot pair selector (ISA A/B layout)
  int l15  = lane & 15;

  int mb0 = blockIdx.x * 128;
  int nb0 = blockIdx.y * 64;
  int arow = mb0 + w * 16 + l15;

  const float* __restrict__ Ar = A + (long long)arow * K;
  __builtin_prefetch(Ar, 0, 0);              // global_prefetch_b8

  int P = (K + KBLK - 1) / KBLK;             // number of k panels

#ifdef HAVE_TDM
  if (w == 0) {
    int r0 = (K < KBLK) ? K : KBLK;
    tdm_load_panel(B + (long long)0 * Nc + nb0, (unsigned)(uintptr_t)&Bs[0][0], r0, Nc);
  }
#endif

  v8f acc0 = {}, acc1 = {}, acc2 = {}, acc3 = {};

  for (int p = 0; p < P; ++p) {
    int k0 = p * KBLK;
    int kblk = K - k0; if (kblk > KBLK) kblk = KBLK;
    int cur = p & 1;

    __syncthreads();                         // prev readers of Bs[1-cur] are done
#ifdef HAVE_TDM
    if (p + 1 < P) {
      if (w == 0) {
        int k1 = k0 + KBLK;
        int r1 = K - k1; if (r1 > KBLK) r1 = KBLK;
        tdm_load_panel(B + (long long)k1 * Nc + nb0,
                       (unsigned)(uintptr_t)&Bs[1 - cur][0], r1, Nc);
        __builtin_amdgcn_s_wait_tensorcnt(1);   // panel p complete (in-order TDM)
      }
    } else {
      if (w == 0) __builtin_amdgcn_s_wait_tensorcnt(0);
    }
#else
    for (int t = threadIdx.x; t < kblk * 64; t += 256) {   // sync fallback
      int kk = t >> 6, cc = t & 63;
      Bs[cur][kk * 64 + cc] = B[(long long)(k0 + kk) * Nc + nb0 + cc];
    }
#endif
    __syncthreads();                         // panel p visible to all waves

    const float* __restrict__ Bp = &Bs[cur][0];
    for (int kk = 0; kk < kblk; kk += 4) {
      int ka = kk + (half << 1);             // even
      v2f a = *(const v2f*)(Ar + k0 + ka);   // global_load_b64 (A fragment)
      v2f b0, b1, b2, b3;                    // ds_load_b32 pairs from row-major panel
      b0.x = Bp[ka * 64 + l15];        b0.y = Bp[(ka + 1) * 64 + l15];
      b1.x = Bp[ka * 64 + 16 + l15];   b1.y = Bp[(ka + 1) * 64 + 16 + l15];
      b2.x = Bp[ka * 64 + 32 + l15];   b2.y = Bp[(ka + 1) * 64 + 32 + l15];
      b3.x = Bp[ka * 64 + 48 + l15];   b3.y = Bp[(ka + 1) * 64 + 48 + l15];
      acc0 = __builtin_amdgcn_wmma_f32_16x16x4_f32(false, a, false, b0, (short)0, acc0, false, false);
      acc1 = __builtin_amdgcn_wmma_f32_16x16x4_f32(false, a, false, b1, (short)0, acc1, false, false);
      acc2 = __builtin_amdgcn_wmma_f32_16x16x4_f32(false, a, false, b2, (short)0, acc2, false, false);
      acc3 = __builtin_amdgcn_wmma_f32_16x16x4_f32(false, a, false, b3, (short)0, acc3, false, false);
    }
  }

  v8f accs[4] = {acc0, acc1, acc2, acc3};
#pragma unroll
  for (int nt = 0; nt < 4; ++nt) {
    int col = nb0 + nt * 16 + l15;
    float bv = bias[col];
#pragma unroll
    for (int r = 0; r < 8; ++r) {
      int row = mb0 + w * 16 + r + (half << 3);
      float v = accs[nt][r] + bv;
      if (relu) v = fmaxf(v, 0.0f);
      C[(long long)row * Nc + col] = v;
    }
  }
}

// ---------------- simple one-tile-per-wave WMMA GEMM (for Nc==16 readout) ----------
__global__ void k_wmma_gemm_simple(const float* __restrict__ A, const float* __restrict__ B,
                                   const float* __restrict__ bias, float* __restrict__ C,
                                   int M, int K, int Nc, int relu) {
  int wave = (int)((blockIdx.x * blockDim.x + threadIdx.x) >> 5);
  int lane = threadIdx.x & 31;
  int tilesN = Nc >> 4;
  int totTiles = (M >> 4) * tilesN;
  if (wave >= totTiles) return;
  int tm = wave / tilesN, tn = wave - tm * tilesN;
  int m0 = tm << 4, n0 = tn << 4;
  int half = lane >> 4, l15 = lane & 15;
  int arow = m0 + l15, bcol = n0 + l15;
  const float* __restrict__ Ar = A + (long long)arow * K;
  v8f acc = {};
  for (int k = 0; k < K; k += 4) {
    int ka = k + (half << 1);
    v2f a = *(const v2f*)(Ar + ka);
    v2f b;
    b.x = B[(long long)ka * Nc + bcol];
    b.y = B[(long long)(ka + 1) * Nc + bcol];
    acc = __builtin_amdgcn_wmma_f32_16x16x4_f32(false, a, false, b, (short)0, acc, false, false);
  }
  float bv = bias[bcol];
#pragma unroll
  for (int r = 0; r < 8; ++r) {
    int row = m0 + r + (half << 3);
    float v = acc[r] + bv;
    if (relu) v = fmaxf(v, 0.0f);
    C[(long long)row * Nc + bcol] = v;
  }
}

// ---------------- host-side orchestration ----------------

static inline void launch_gemm(const float* A, const float* B, const float* bias, float* C,
                               int M, int K, int Nc, int relu, hipStream_t s) {
  dim3 grid(M / 128, Nc / 64);
  k_wmma_gemm_lds<<<grid, 256, 0, s>>>(A, B, bias, C, M, K, Nc, relu);
}

static inline void launch_gather(const float* x, const int* row_start, const int* nbr,
                                 const float* ns, const float* nd, float* agg,
                                 int din, hipStream_t s) {
  long long total = (long long)NN * din;
  int blocks = (int)((total + 255) / 256);
  k_gather<<<blocks, 256, 0, s>>>(x, row_start, nbr, ns, nd, agg, din, total);
}

extern "C" void kernel_launch(void* const* d_in, const int* in_sizes, int n_in,
                              void* d_out, int out_size, void* d_ws, size_t ws_size,
                              hipStream_t stream) {
  const float* raw  = (const float*)d_in[0];
  const int*   src  = (const int*)  d_in[1];
  const int*   dst  = (const int*)  d_in[2];
  const float* eps  = (const float*)d_in[3];
  const float* W1   = (const float*)d_in[4];   const float* b1   = (const float*)d_in[5];
  const float* W2   = (const float*)d_in[6];   const float* b2   = (const float*)d_in[7];
  const float* W3   = (const float*)d_in[8];   const float* b3   = (const float*)d_in[9];
  const float* W4   = (const float*)d_in[10];  const float* b4   = (const float*)d_in[11];
  const float* Wmu  = (const float*)d_in[12];  const float* bmu  = (const float*)d_in[13];
  const float* Wlv  = (const float*)d_in[14];  const float* blv  = (const float*)d_in[15];
  const float* Wdec = (const float*)d_in[16];  const float* bdec = (const float*)d_in[17];
  const float* Wg1  = (const float*)d_in[18];  const float* bg1  = (const float*)d_in[19];
  const float* Wg2  = (const float*)d_in[20];  const float* bg2  = (const float*)d_in[21];
  const float* Wout = (const float*)d_in[22];  const float* bout = (const float*)d_in[23];

  // Outputs: [recon(N*16) | mu(256*128) | logvar(256*128)]
  float* recon = (float*)d_out;
  float* mu    = recon + (long long)NN * IN_FEATS;
  float* lv    = mu + (long long)N_GRAPHS * Z_DIM;

  // Workspace carve-up: int region first, then floats
  int* iws = (int*)d_ws;
  int* deg_src   = iws;                 // NN
  int* deg_dst   = deg_src + NN;        // NN
  int* cursor    = deg_dst + NN;        // NN
  int* row_start = cursor + NN;         // NN+1
  int* nbr       = row_start + NN + 1;  // EE (src node ids, grouped by dst)

  float* fws = (float*)(nbr + EE + 63);
  float* ns     = fws;                              // NN
  float* nd     = ns + NN;                          // NN
  float* h1     = nd + NN;                          // NN*HIDDEN
  float* h2     = h1 + (long long)NN * HIDDEN;
  float* h3     = h2 + (long long)NN * HIDDEN;
  float* h4     = h3 + (long long)NN * HIDDEN;
  float* agg    = h4 + (long long)NN * HIDDEN;      // NN*DEC_IN (max din)
  float* z      = agg + (long long)NN * DEC_IN;     // 256*128
  float* latent = z + (long long)N_GRAPHS * Z_DIM;  // NN*HIDDEN
  float* x5     = latent + (long long)NN * HIDDEN;  // NN*HIDDEN
  float* x6     = x5 + (long long)NN * HIDDEN;      // NN*HIDDEN

  // 1) CSR build + symmetric-norm factors (once; 1M placement atomics total)
  k_zero_i<<<(3 * NN + 255) / 256, 256, 0, stream>>>(deg_src, 3LL * NN);
  k_degree<<<(EE + 255) / 256, 256, 0, stream>>>(src, dst, deg_src, deg_dst);
  k_finalize_norm<<<(NN + 255) / 256, 256, 0, stream>>>(deg_src, deg_dst, ns, nd);
  k_scan<<<1, 1024, 0, stream>>>(deg_dst, row_start);
  k_fill<<<(EE + 255) / 256, 256, 0, stream>>>(src, dst, row_start, cursor, nbr);

  // 2) encoder GraphConvs
  launch_gather(raw, row_start, nbr, ns, nd, agg, IN_FEATS, stream);
  launch_gemm(agg, W1, b1, h1, NN, IN_FEATS, HIDDEN, 1, stream);

  launch_gather(h1, row_start, nbr, ns, nd, agg, HIDDEN, stream);
  launch_gemm(agg, W2, b2, h2, NN, HIDDEN, HIDDEN, 1, stream);

  launch_gather(h2, row_start, nbr, ns, nd, agg, HIDDEN, stream);
  launch_gemm(agg, W3, b3, h3, NN, HIDDEN, HIDDEN, 1, stream);

  launch_gather(h3, row_start, nbr, ns, nd, agg, HIDDEN, stream);
  launch_gemm(agg, W4, b4, h4, NN, HIDDEN, HIDDEN, 1, stream);

  // 3) latent heads on h_flat = h4 viewed [256, 16384]
  launch_gemm(h4, Wmu, bmu, mu, N_GRAPHS, N_PER * HIDDEN, Z_DIM, 0, stream);
  launch_gemm(h4, Wlv, blv, lv, N_GRAPHS, N_PER * HIDDEN, Z_DIM, 0, stream);

  // 4) reparameterize + decode
  k_reparam<<<(N_GRAPHS * Z_DIM + 255) / 256, 256, 0, stream>>>(mu, lv, eps, z);
  launch_gemm(z, Wdec, bdec, latent, N_GRAPHS, Z_DIM, N_PER * HIDDEN, 0, stream);

  // 5) decoder GraphConv 1 with fused concat-gather
  {
    long long total = (long long)NN * DEC_IN;
    k_gather_concat<<<(int)((total + 255) / 256), 256, 0, stream>>>(
        latent, raw, h1, h3, row_start, nbr, ns, nd, agg, total);
  }
  launch_gemm(agg, Wg1, bg1, x5, NN, DEC_IN, HIDDEN, 1, stream);

  // 6) decoder GraphConv 2
  launch_gather(x5, row_start, nbr, ns, nd, agg, HIDDEN, stream);
  launch_gemm(agg, Wg2, bg2, x6, NN, HIDDEN, HIDDEN, 1, stream);

  // 7) readout (Nc=16 -> simple kernel) + atomic-number patch
  k_wmma_gemm_simple<<<(NN / 16 * 32 + 255) / 256, 256, 0, stream>>>(
      x6, Wout, bout, recon, NN, HIDDEN, IN_FEATS, 0);
  k_fixcol0<<<(NN + 255) / 256, 256, 0, stream>>>(raw, recon);
}